// BasicGraphModel_23038204575791
// MI455X (gfx1250) — compile-verified
//
#include <hip/hip_runtime.h>

#define N_NODES 100000

typedef __attribute__((ext_vector_type(2))) float v2f;
typedef __attribute__((ext_vector_type(8))) float v8f;

// ---------------------------------------------------------------------------
// Degree accumulation: one thread per edge, float atomics into out/in degree.
// ---------------------------------------------------------------------------
__global__ void deg_kernel(const int* __restrict__ src, const int* __restrict__ dst,
                           float* __restrict__ odeg, float* __restrict__ ideg, int E) {
    int e = blockIdx.x * blockDim.x + threadIdx.x;
    if (e < E) {
        atomicAdd(&odeg[src[e]], 1.0f);
        atomicAdd(&ideg[dst[e]], 1.0f);
    }
}

// In-place deg -> rsqrt(max(deg, 1))
__global__ void norm_kernel(float* __restrict__ deg, int n) {
    int i = blockIdx.x * blockDim.x + threadIdx.x;
    if (i < n) {
        deg[i] = rsqrtf(fmaxf(deg[i], 1.0f));
    }
}

// ---------------------------------------------------------------------------
// Fused Y = (X * ns[:,None]) @ W using V_WMMA_F32_16X16X4_F32 (fp32 precision).
// One wave computes one 16x16 tile of Y; block = (DOUT/16) waves covering all
// column tiles of a 16-row stripe. grid.x = N/16 row stripes (N % 16 == 0).
//
// A (16x4 f32) VGPR layout: lanes 0-15 -> M=lane, regs {K=0,K=1};
//                           lanes 16-31 -> M=lane-16, regs {K=2,K=3}.
// B (4x16 f32): mirrored    lanes 0-15 -> N=lane, regs {K=0,K=1};
//                           lanes 16-31 -> N=lane-16, regs {K=2,K=3}.
// C/D (16x16 f32): reg v, lanes 0-15 -> (M=v, N=lane); lanes 16-31 -> (M=v+8).
// ---------------------------------------------------------------------------
template <int DIN, int DOUT>
__global__ void gemm_norm_wmma(const float* __restrict__ X, const float* __restrict__ ns,
                               const float* __restrict__ W, float* __restrict__ Y) {
    const int wave = threadIdx.x >> 5;
    const int lane = threadIdx.x & 31;
    const int hl   = lane & 15;
    const bool hi  = (lane >= 16);

    const int row0 = blockIdx.x * 16;
    const int col0 = wave * 16;
    const int arow = row0 + hl;     // A row for this lane
    const int col  = col0 + hl;     // B/C column for this lane

    const float nsr = ns[arow];
    const float* __restrict__ xrow = X + (size_t)arow * DIN;

    v8f acc = {};
#pragma unroll 8
    for (int k = 0; k < DIN; k += 4) {
        const int ka = k + (hi ? 2 : 0);
        v2f a, b;
        a[0] = xrow[ka] * nsr;
        a[1] = xrow[ka + 1] * nsr;
        b[0] = W[(size_t)ka * DOUT + col];
        b[1] = W[(size_t)(ka + 1) * DOUT + col];
        acc = __builtin_amdgcn_wmma_f32_16x16x4_f32(
            /*neg_a=*/false, a, /*neg_b=*/false, b,
            /*c_mod=*/(short)0, acc, /*reuse_a=*/false, /*reuse_b=*/false);
    }

    const int crow0 = row0 + (hi ? 8 : 0);
#pragma unroll
    for (int v = 0; v < 8; ++v) {
        Y[(size_t)(crow0 + v) * DOUT + col] = acc[v];
    }
}

// ---------------------------------------------------------------------------
// Edge scatter: M[dst[e]] += H[src[e]].  One thread owns 4 consecutive
// features of one edge (float4 gather, 4 float atomics). D = 1 << (P+2).
// blockDim.x = 256 -> (256 >> P) edges per block.
// ---------------------------------------------------------------------------
template <int D, int P>
__global__ void scatter_kernel(const float* __restrict__ H, const int* __restrict__ src,
                               const int* __restrict__ dst, float* __restrict__ M, int E) {
    constexpr int PER_EDGE = D / 4;             // == 1 << P
    const int e = blockIdx.x * (256 >> P) + (threadIdx.x >> P);
    const int f = (threadIdx.x & (PER_EDGE - 1)) << 2;
    if (e >= E) return;

    const int s = src[e];
    const int d = dst[e];
    const float4 v = *reinterpret_cast<const float4*>(H + (size_t)s * D + f);
    float* mp = M + (size_t)d * D + f;
    atomicAdd(mp + 0, v.x);
    atomicAdd(mp + 1, v.y);
    atomicAdd(mp + 2, v.z);
    atomicAdd(mp + 3, v.w);
}

// ---------------------------------------------------------------------------
// out = m * norm_dst[:,None] + b (+ optional ReLU), elementwise over N*D.
// ---------------------------------------------------------------------------
template <int D, bool RELU>
__global__ void finalize_kernel(const float* __restrict__ M, const float* __restrict__ nd,
                                const float* __restrict__ b, float* __restrict__ O, int total) {
    int idx = blockIdx.x * blockDim.x + threadIdx.x;
    if (idx >= total) return;
    const int row = idx / D;          // D is a power of two -> shift
    const int col = idx & (D - 1);
    float v = M[idx] * nd[row] + b[col];
    if (RELU) v = fmaxf(v, 0.0f);
    O[idx] = v;
}

// ---------------------------------------------------------------------------
extern "C" void kernel_launch(void* const* d_in, const int* in_sizes, int n_in,
                              void* d_out, int out_size, void* d_ws, size_t ws_size,
                              hipStream_t stream) {
    (void)n_in; (void)out_size; (void)ws_size;

    const float* x  = (const float*)d_in[0];
    const int*   src = (const int*)d_in[1];
    const int*   dst = (const int*)d_in[2];
    const float* W1 = (const float*)d_in[3];
    const float* b1 = (const float*)d_in[4];
    const float* W2 = (const float*)d_in[5];
    const float* b2 = (const float*)d_in[6];
    const float* W3 = (const float*)d_in[7];
    const float* b3 = (const float*)d_in[8];

    const int E = in_sizes[1];
    const int N = N_NODES;

    // Workspace layout (floats): ns[NPAD] | nd[NPAD] | bufA[N*128] | bufB[N*128]
    const size_t NPAD = 100352;  // N rounded up, keeps buffers 512B-aligned
    float* ws   = (float*)d_ws;
    float* ns   = ws;
    float* nd   = ws + NPAD;
    float* bufA = ws + 2 * NPAD;
    float* bufB = bufA + (size_t)N * 128;

    // --- degrees -> normalization factors (in place) ---
    hipMemsetAsync(ns, 0, 2 * NPAD * sizeof(float), stream);
    deg_kernel<<<(E + 255) / 256, 256, 0, stream>>>(src, dst, ns, nd, E);
    norm_kernel<<<(int)((2 * NPAD + 255) / 256), 256, 0, stream>>>(ns, (int)(2 * NPAD));

    const int ROWT = N / 16;  // 6250 row stripes

    // --- Layer 1: Din=128 -> 128, ReLU ---
    gemm_norm_wmma<128, 128><<<ROWT, 256, 0, stream>>>(x, ns, W1, bufA);
    hipMemsetAsync(bufB, 0, (size_t)N * 128 * sizeof(float), stream);
    scatter_kernel<128, 5><<<(E + 7) / 8, 256, 0, stream>>>(bufA, src, dst, bufB, E);
    finalize_kernel<128, true><<<(N * 128 + 255) / 256, 256, 0, stream>>>(bufB, nd, b1, bufA, N * 128);

    // --- Layer 2: 128 -> 128, ReLU ---
    gemm_norm_wmma<128, 128><<<ROWT, 256, 0, stream>>>(bufA, ns, W2, bufB);
    hipMemsetAsync(bufA, 0, (size_t)N * 128 * sizeof(float), stream);
    scatter_kernel<128, 5><<<(E + 7) / 8, 256, 0, stream>>>(bufB, src, dst, bufA, E);
    finalize_kernel<128, true><<<(N * 128 + 255) / 256, 256, 0, stream>>>(bufA, nd, b2, bufB, N * 128);

    // --- Layer 3: 128 -> 64, no activation, write d_out ---
    gemm_norm_wmma<128, 64><<<ROWT, 128, 0, stream>>>(bufB, ns, W3, bufA);
    hipMemsetAsync(bufB, 0, (size_t)N * 64 * sizeof(float), stream);
    scatter_kernel<64, 4><<<(E + 15) / 16, 256, 0, stream>>>(bufA, src, dst, bufB, E);
    finalize_kernel<64, false><<<(N * 64 + 255) / 256, 256, 0, stream>>>(bufB, nd, b3, (float*)d_out, N * 64);
}